// SoftPERLoss_38259568672846
// MI455X (gfx1250) — compile-verified
//
#include <hip/hip_runtime.h>
#include <hip/hip_bf16.h>

// Problem constants (match reference)
#define T_DIM 1024
#define B_DIM 64
#define C_DIM 512
#define U_MAX 128

static constexpr float TAU_C  = 0.2f;
static constexpr float INVTAU = 5.0f;   // 1/TAU
static constexpr float INS_C  = 1.0f;
static constexpr float DEL_C  = 1.0f;

typedef __attribute__((ext_vector_type(2))) float v2f;
typedef __attribute__((ext_vector_type(8))) float v8f;

__device__ __forceinline__ float softmin3(float a, float b, float c) {
    // -TAU * logsumexp(-x/TAU) == m - TAU*log( sum exp((m-x)/TAU) ), m = min
    float m = fminf(a, fminf(b, c));
    float s = __expf((m - a) * INVTAU) + __expf((m - b) * INVTAU) + __expf((m - c) * INVTAU);
    return m - TAU_C * __logf(s);
}

// One block per batch element. Anti-diagonal wavefront over the (il+1)x(tl+1)
// DP table. Thread u owns column i=u. Per-step LDS traffic minimized to one
// ds_load + one ds_store by register-carrying:
//   vPrev  = diag(d-1)[u]   (this thread's own previous write)
//   p2reg  = diag(d-2)[u-1] (last iteration's left-neighbor LDS read)
// -> only TWO alternating diagonal buffers are needed.
__global__ __launch_bounds__(160)
void softper_dp_kernel(const float* __restrict__ lp,          // (T,B,C) log-probs
                       const int*   __restrict__ targets,     // (B*UMAX,)
                       const int*   __restrict__ input_lengths,
                       const int*   __restrict__ target_lengths,
                       float*       __restrict__ partial)     // (B,) per-batch loss
{
    const int b = blockIdx.x;
    const int u = threadIdx.x;

    __shared__ float diag[2][U_MAX + 4];
    __shared__ int   cls[U_MAX];
    __shared__ int   s_off;

    int il = input_lengths[b];  il = il < 1 ? 1 : (il > T_DIM ? T_DIM : il);
    int tl = target_lengths[b]; tl = tl < 1 ? 1 : (tl > U_MAX ? U_MAX : tl);

    // Exclusive prefix sum of raw target_lengths (64 elems, serial, once).
    if (u == 0) {
        int off = 0;
        for (int k = 0; k < b; ++k) off += target_lengths[k];
        s_off = off;
    }
    __syncthreads();

    if (u < U_MAX) {
        int idx = s_off + u;
        const int lim = B_DIM * U_MAX - 1;
        idx = idx < lim ? idx : lim;     // matches reference clip
        cls[u] = targets[idx];           // only consumed for u < tl
    }
    if (u == 0) {
        diag[1][0] = INS_C;  // D[1][0] = 1*INS  (diagonal d=1, entry 0)
        diag[1][1] = DEL_C;  // D[0][1] = 1*DEL  (diagonal d=1, entry 1)
    }
    __syncthreads();

    const bool isCell = (u >= 1) && (u <= tl);
    const int  myCls  = isCell ? cls[u - 1] : 0;
    const float* gbase   = lp + (size_t)b * C_DIM + myCls;   // + t * (B*C)
    const size_t gstride = (size_t)B_DIM * C_DIM;

    // Gather log_prob for cell (j=d-u, i=u): t = j-1 = d-u-1 (clamped; value
    // only consumed when the cell is valid).
    auto gload = [&](int d) -> float {
        int t = d - u - 1;
        t = t < 0 ? 0 : (t >= T_DIM ? T_DIM - 1 : t);
        return gbase[(size_t)t * gstride];
    };

    // Depth-3 software prefetch pipeline (addresses advance by gstride/step).
    float gA = isCell ? gload(2) : 0.0f;   // for diagonal d
    float gB = isCell ? gload(3) : 0.0f;   // d+1
    float gC = isCell ? gload(4) : 0.0f;   // d+2

    const int dEnd = il + tl;
    float result = 0.0f;

    // Register-carried DP state.
    float vPrev = (float)u * DEL_C;  // diag(d-1)[u]; first consumed at d=u+1
                                     // where it must be u*DEL (top boundary)
    float p2reg = 0.0f;              // diag(d-2)[u-1]; u=1 first consumes
                                     // D[0][0]=0 at d=2

    for (int d = 2; d <= dEnd; ++d) {
        // Issue gather for diagonal d+3 now (hidden behind this step).
        float gNew = isCell ? gload(d + 3) : 0.0f;

        // Staggered speculative prefetch to pull the log_prob stream into L2
        // ahead of the wavefront (global_prefetch_b8).
        if (isCell && (((d + u) & 7) == 0)) {
            int tp = d - u - 1 + 32;
            if (tp < il) __builtin_prefetch(gbase + (size_t)tp * gstride, 0, 0);
        }

        float*       cur = diag[d & 1];        // overwrites diagonal d-2 (its
        const float* p1  = diag[(d & 1) ^ 1];  // LDS readers ran at step d-1)

        __syncthreads();  // step d-1 writes visible; its reads done before ours

        // Left-neighbor value on diagonal d-1 (only LDS load of the step).
        // Read unconditionally so p2reg is primed before the first compute.
        float left = isCell ? p1[u - 1] : 0.0f;

        // Boundary cells of diagonal d (stored for the left-neighbor read).
        if (u == 0) cur[0] = (float)d * INS_C;                // D[d][0]
        if (u == d && u <= U_MAX) {
            cur[u] = (float)u * DEL_C;                        // D[0][d]
            vPrev  = (float)u * DEL_C;                        // == init value
        }

        const int j = d - u;
        if (isCell && j >= 1 && j <= il) {
            float sub = 1.0f - __expf(gA);                    // 1 - p_corr
            float v = softmin3(left  + DEL_C,                 // D[j][i-1] + DEL
                               vPrev + INS_C,                 // D[j-1][i] + INS
                               p2reg + sub);                  // D[j-1][i-1]+sub
            cur[u] = v;
            vPrev  = v;
            if (j == il && u == tl) result = v;               // final cell
        }
        p2reg = left;              // becomes diag(d-1)[u-1] for step d+1
        gA = gB; gB = gC; gC = gNew;
    }

    if (u == tl) {
        partial[b] = result / ((float)tl + 1e-8f);
    }
}

// Deterministic 64 -> 1 mean via one V_WMMA_F32_16X16X4_F32.
// A = ones(16x4), B(4x16) packed bijectively with the 64 per-batch losses:
// D[0][c] = sum_k B[k][c]; summing the 16 column sums in fixed order gives the
// exact, order-fixed total regardless of the lane->(k,c) mapping details.
// Single wave32, EXEC all ones at the WMMA (required by ISA).
__global__ __launch_bounds__(32)
void reduce_mean_wmma(const float* __restrict__ part, float* __restrict__ out)
{
    const int l = threadIdx.x;           // 0..31
    const int c  = l & 15;               // column 0..15
    const int kh = (l >> 4) * 2;         // K base: lanes 0-15 -> K{0,1}, 16-31 -> K{2,3}

    v2f a;  a.x = 1.0f;  a.y = 1.0f;     // A-matrix: all ones
    v2f bv;                              // B[k][c] per 32-bit B striping
    bv.x = part[c * 4 + kh];
    bv.y = part[c * 4 + kh + 1];
    v8f cz = {};

    v8f d = __builtin_amdgcn_wmma_f32_16x16x4_f32(
        /*neg_a=*/false, a, /*neg_b=*/false, bv,
        /*c_mod=*/(short)0, cz, /*reuse_a=*/false, /*reuse_b=*/false);

    __shared__ float s[16];
    if (l < 16) s[l] = d[0];             // row M=0: D[0][c] in VGPR0, lanes 0-15
    __syncthreads();
    if (l == 0) {
        float acc = 0.0f;
        for (int i = 0; i < 16; ++i) acc += s[i];
        out[0] = acc * (1.0f / (float)B_DIM);
    }
}

extern "C" void kernel_launch(void* const* d_in, const int* in_sizes, int n_in,
                              void* d_out, int out_size, void* d_ws, size_t ws_size,
                              hipStream_t stream) {
    const float* lp       = (const float*)d_in[0];  // log_probs (T,B,C) f32
    const int*   targets  = (const int*)  d_in[1];  // (B*UMAX,) i32
    const int*   ilen     = (const int*)  d_in[2];  // (B,) i32
    const int*   tlen     = (const int*)  d_in[3];  // (B,) i32
    float*       partial  = (float*)d_ws;           // 64 floats scratch
    float*       out      = (float*)d_out;          // scalar f32

    softper_dp_kernel<<<B_DIM, 160, 0, stream>>>(lp, targets, ilen, tlen, partial);
    reduce_mean_wmma<<<1, 32, 0, stream>>>(partial, out);
}